// VectorQuantGroup_60455959658949
// MI455X (gfx1250) — compile-verified
//
#include <hip/hip_runtime.h>
#include <math.h>

// ---------------------------------------------------------------------------
// VectorQuantGroup for MI455X (gfx1250, wave32)
// x: (16,4096,1,128) f32 -> B=65536 rows, D=128.  emb: (1,512,128) f32, K=512.
// Heavy op: fp32 distance GEMM via V_WMMA_F32_16X16X4_F32 (bf16 WMMA would
// break class ranking: deltas ~1e-3 on top of ||x||^2 ~ 128).
// v2: 32 rows per wave (2 A panels / 2 acc chains) -> each B float2 feeds two
// WMMAs: halves L2 traffic, doubles wmma:vmem ratio, adds XDL ILP.
// ---------------------------------------------------------------------------

#define BROWS   65536
#define DIMS    128
#define KCLS    512
#define NGROUP  16
#define CPG     32
#define NSAMP   8

typedef __attribute__((ext_vector_type(2))) float v2f;
typedef __attribute__((ext_vector_type(8))) float v8f;

// ---------------------------------------------------------------------------
// Kernel A: build B-panel layout of emb, squared norms, zero histogram.
// embB element (float2) index p = (nt*32 + c)*32 + lane holds, for j in {0,1}:
//   emb[class][k] with class = nt*16 + (lane&15), k = 4c + 2*(lane>>4) + j
// which is exactly the B-operand (4x16 f32) VGPR pair for chunk c of tile nt.
// ---------------------------------------------------------------------------
__global__ __launch_bounds__(256) void vq_prep_kernel(
    const float* __restrict__ emb, float* __restrict__ embB,
    float* __restrict__ e2, int* __restrict__ hist)
{
    const int t = threadIdx.x;
    hist[t] = 0;
    hist[t + 256] = 0;

    for (int k = t; k < KCLS; k += 256) {
        float s = 0.f;
        for (int dd = 0; dd < DIMS; ++dd) {
            float e = emb[k * DIMS + dd];
            s += e * e;
        }
        e2[k] = s;
    }

    float2* embB2 = (float2*)embB;
    for (int p = t; p < (KCLS / 16) * 32 * 32; p += 256) {  // 32768 float2
        const int nt = p >> 10;
        const int c  = (p >> 5) & 31;
        const int l  = p & 31;
        const int cls = nt * 16 + (l & 15);
        const int k0  = 4 * c + 2 * (l >> 4);
        float2 v;
        v.x = emb[cls * DIMS + k0];
        v.y = emb[cls * DIMS + k0 + 1];
        embB2[p] = v;
    }
}

// ---------------------------------------------------------------------------
// Kernel B: distance GEMM.  One wave handles 32 rows x all 512 classes.
// A (16x4 f32) layout: lane l holds row (l&15), K = 4c + 2*(l>>4) + j.
// C (16x16 f32) layout: VGPR v, lane l -> row v + 8*(l>>4), class (l&15).
// Two A panels (rows 0-15 and 16-31 of the block) share every B load.
// dws is chunk-local: dws[localrow*512 + class] = sqrt(max(d2,0)).
// ---------------------------------------------------------------------------
__global__ __launch_bounds__(128) void vq_gemm_kernel(
    const float* __restrict__ x, const float* __restrict__ embB,
    const float* __restrict__ e2, float* __restrict__ dws, int rowStart)
{
    const int lane  = threadIdx.x & 31;
    const int wave  = blockIdx.x * (blockDim.x >> 5) + (threadIdx.x >> 5);
    const int rowBlock = wave * 32;           // chunk-local base row
    const int half  = lane >> 4;
    const int rlane = lane & 15;

    // ---- load two x row panels into A-layout registers + ||x||^2 ----
    const float4* xr0 =
        (const float4*)(x + (size_t)(rowStart + rowBlock + rlane) * DIMS);
    const float4* xr1 =
        (const float4*)(x + (size_t)(rowStart + rowBlock + 16 + rlane) * DIMS);
    float xa0[64], xa1[64];
    float x2a = 0.f, x2b = 0.f;
#pragma unroll
    for (int c = 0; c < 32; ++c) {
        float4 q0 = xr0[c];
        float4 q1 = xr1[c];
        float a0 = half ? q0.z : q0.x;
        float a1 = half ? q0.w : q0.y;
        float b0 = half ? q1.z : q1.x;
        float b1 = half ? q1.w : q1.y;
        xa0[2 * c]     = a0;  xa0[2 * c + 1] = a1;
        xa1[2 * c]     = b0;  xa1[2 * c + 1] = b1;
        x2a += a0 * a0 + a1 * a1;
        x2b += b0 * b0 + b1 * b1;
    }
    x2a += __shfl_xor(x2a, 16, 32);           // lanes l, l+16 cover full row
    x2b += __shfl_xor(x2b, 16, 32);
    // x2 for C-layout row (v + 8*half), per accumulator VGPR v
    float x2s0[8], x2s1[8];
#pragma unroll
    for (int v = 0; v < 8; ++v) {
        x2s0[v] = __shfl(x2a, v + 8 * half, 32);
        x2s1[v] = __shfl(x2b, v + 8 * half, 32);
    }

    const float2* embB2 = (const float2*)embB;

#pragma unroll 1
    for (int nt = 0; nt < KCLS / 16; ++nt) {  // 32 class tiles
        v8f acc0 = {0.f, 0.f, 0.f, 0.f, 0.f, 0.f, 0.f, 0.f};
        v8f acc1 = {0.f, 0.f, 0.f, 0.f, 0.f, 0.f, 0.f, 0.f};
        const float e2v = e2[nt * 16 + rlane];
        const float2* bp = embB2 + (size_t)nt * 32 * 32 + lane;
#pragma unroll
        for (int c = 0; c < 32; ++c) {        // K = 128 in steps of 4
            float2 bb = bp[c * 32];
            v2f b;  b.x = bb.x;       b.y = bb.y;
            v2f a0; a0.x = xa0[2 * c]; a0.y = xa0[2 * c + 1];
            v2f a1; a1.x = xa1[2 * c]; a1.y = xa1[2 * c + 1];
            acc0 = __builtin_amdgcn_wmma_f32_16x16x4_f32(
                false, a0, false, b, (short)0, acc0, false, false);
            acc1 = __builtin_amdgcn_wmma_f32_16x16x4_f32(
                false, a1, false, b, (short)0, acc1, false, false);
        }
#pragma unroll
        for (int v = 0; v < 8; ++v) {
            float d20 = x2s0[v] + e2v - 2.0f * acc0[v];
            float d21 = x2s1[v] + e2v - 2.0f * acc1[v];
            float dd0 = sqrtf(fmaxf(d20, 0.f));
            float dd1 = sqrtf(fmaxf(d21, 0.f));
            size_t r0 = (size_t)(rowBlock + v + 8 * half);
            size_t r1 = r0 + 16;
            dws[r0 * KCLS + nt * 16 + rlane] = dd0;
            dws[r1 * KCLS + nt * 16 + rlane] = dd1;
        }
    }
}

// ---------------------------------------------------------------------------
// Kernel C: per-row finalize.  One wave per row.
// lane l holds d[32*i + l] for i = 0..15  ->  i IS the group index.
// ---------------------------------------------------------------------------
__global__ __launch_bounds__(256) void vq_finalize_kernel(
    const float* __restrict__ x, const float* __restrict__ emb,
    const float* __restrict__ dws, float* __restrict__ out0,
    float* __restrict__ out1, float* __restrict__ out2,
    int* __restrict__ hist, int rowStart)
{
    const int lane = threadIdx.x & 31;
    const int lrow = blockIdx.x * (blockDim.x >> 5) + (threadIdx.x >> 5);
    const int R    = rowStart + lrow;

    const float* drow = dws + (size_t)lrow * KCLS;
    float v[NGROUP];
#pragma unroll
    for (int i = 0; i < NGROUP; ++i) v[i] = drow[i * 32 + lane];

    // group sums (mean argmin == sum argmin); every lane ends with all sums
    float gs[NGROUP];
#pragma unroll
    for (int i = 0; i < NGROUP; ++i) {
        float s = v[i];
        s += __shfl_xor(s, 16, 32);
        s += __shfl_xor(s, 8, 32);
        s += __shfl_xor(s, 4, 32);
        s += __shfl_xor(s, 2, 32);
        s += __shfl_xor(s, 1, 32);
        gs[i] = s;
    }
    int gbest = 0;
    float best = gs[0];
#pragma unroll
    for (int i = 1; i < NGROUP; ++i) {
        if (gs[i] < best) { best = gs[i]; gbest = i; }   // first-min, as argmin
    }

    // this lane's distance within the winning group
    float cur = v[0];
#pragma unroll
    for (int i = 1; i < NGROUP; ++i) cur = (i == gbest) ? v[i] : cur;

    // top-8 smallest d (== top-8 largest 1/d), stable: ties -> lower class
    float w[NSAMP];
    int   cls[NSAMP];
    float wsum = 0.f;
#pragma unroll
    for (int s = 0; s < NSAMP; ++s) {
        unsigned long long key =
            ((unsigned long long)__float_as_uint(cur) << 32) | (unsigned)lane;
        {
            unsigned long long o;
            o = __shfl_xor(key, 16, 32); key = (o < key) ? o : key;
            o = __shfl_xor(key, 8, 32);  key = (o < key) ? o : key;
            o = __shfl_xor(key, 4, 32);  key = (o < key) ? o : key;
            o = __shfl_xor(key, 2, 32);  key = (o < key) ? o : key;
            o = __shfl_xor(key, 1, 32);  key = (o < key) ? o : key;
        }
        const int   mlane = (int)(key & 0xffffffffull);
        const float mval  = __uint_as_float((unsigned)(key >> 32));
        w[s]   = 1.0f / mval;
        cls[s] = gbest * CPG + mlane;
        wsum  += w[s];
        if (lane == mlane) cur = __builtin_inff();       // remove and repeat
    }
    const float inv = 1.0f / wsum;

    // weighted codeword sum; lane handles dims 4*lane .. 4*lane+3
    float4 o = {0.f, 0.f, 0.f, 0.f};
#pragma unroll
    for (int s = 0; s < NSAMP; ++s) {
        const float4 e = *(const float4*)(emb + (size_t)cls[s] * DIMS + lane * 4);
        const float ww = w[s] * inv;
        o.x += ww * e.x; o.y += ww * e.y; o.z += ww * e.z; o.w += ww * e.w;
    }
    const float4 xv = *(const float4*)(x + (size_t)R * DIMS + lane * 4);
    *(float4*)(out0 + (size_t)R * DIMS + lane * 4) = o;   // out0 == output (STE)

    float dx = xv.x - o.x, dy = xv.y - o.y, dz = xv.z - o.z, dw = xv.w - o.w;
    float ss = dx * dx + dy * dy + dz * dz + dw * dw;
    ss += __shfl_xor(ss, 16, 32);
    ss += __shfl_xor(ss, 8, 32);
    ss += __shfl_xor(ss, 4, 32);
    ss += __shfl_xor(ss, 2, 32);
    ss += __shfl_xor(ss, 1, 32);
    if (lane == 0) {
        out1[R] = ss;
        out2[R] = ss;                 // x == x0 -> second term of out2 is 0
        atomicAdd(&hist[cls[0]], 1);  // top-1 code usage
    }
}

// ---------------------------------------------------------------------------
// Kernel D: entropy of top-1 code histogram.
// ---------------------------------------------------------------------------
__global__ __launch_bounds__(32) void vq_entropy_kernel(
    const int* __restrict__ hist, float* __restrict__ out_e)
{
    const int lane = threadIdx.x;
    float acc = 0.f;
    for (int t = 0; t < KCLS / 32; ++t) {
        int h = hist[t * 32 + lane];
        if (h > 0) {
            float pr = (float)h * (1.0f / (float)BROWS);
            acc -= pr * logf(pr);
        }
    }
    acc += __shfl_xor(acc, 16, 32);
    acc += __shfl_xor(acc, 8, 32);
    acc += __shfl_xor(acc, 4, 32);
    acc += __shfl_xor(acc, 2, 32);
    acc += __shfl_xor(acc, 1, 32);
    if (lane == 0) *out_e = acc;
}

// ---------------------------------------------------------------------------
// Host launcher.  Workspace layout:
//   [embB: 256KB][e2: 2KB][hist: 2KB][d-buffer: chunkRows*512*4B]
// Chunked over rows so d-buffer adapts to ws_size (distances stay L2-hot).
// ---------------------------------------------------------------------------
extern "C" void kernel_launch(void* const* d_in, const int* in_sizes, int n_in,
                              void* d_out, int out_size, void* d_ws, size_t ws_size,
                              hipStream_t stream)
{
    const float* x   = (const float*)d_in[0];   // (16,4096,1,128)
    const float* emb = (const float*)d_in[1];   // (1,512,128)

    float* out  = (float*)d_out;
    float* out0 = out;                           // 8388608 floats
    float* out1 = out + 8388608;                 // 65536
    float* out2 = out + 8454144;                 // 65536
    float* oute = out + 8519680;                 // 1

    char*  ws   = (char*)d_ws;
    float* embB = (float*)ws;                            // 262144 B
    float* e2   = (float*)(ws + 262144);                 // 2048 B
    int*   hist = (int*)(ws + 262144 + 2048);            // 2048 B
    float* dbuf = (float*)(ws + 262144 + 4096);

    const size_t fixedBytes = 262144 + 4096;
    size_t avail = (ws_size > fixedBytes) ? (ws_size - fixedBytes) : 0;
    long maxRows = (long)(avail / (KCLS * sizeof(float)));
    long chunk = maxRows - (maxRows % 128);
    if (chunk > BROWS) chunk = BROWS;
    if (chunk < 128)   chunk = 128;   // minimal footprint fallback

    vq_prep_kernel<<<1, 256, 0, stream>>>(emb, embB, e2, hist);

    for (long start = 0; start < BROWS; start += chunk) {
        long rows = BROWS - start;
        if (rows > chunk) rows = chunk;
        // GEMM: 1 wave / 32 rows, 4 waves / block
        vq_gemm_kernel<<<(int)(rows / 128), 128, 0, stream>>>(
            x, embB, e2, dbuf, (int)start);
        // finalize: 1 wave / row, 8 waves / block
        vq_finalize_kernel<<<(int)(rows / 8), 256, 0, stream>>>(
            x, emb, dbuf, out0, out1, out2, hist, (int)start);
    }

    vq_entropy_kernel<<<1, 32, 0, stream>>>(hist, oute);
}